// MultiHeadAttention_66812511257255
// MI455X (gfx1250) — compile-verified
//
#include <hip/hip_runtime.h>
#include <math.h>

typedef float v2f __attribute__((ext_vector_type(2)));
typedef float v4f __attribute__((ext_vector_type(4)));
typedef float v8f __attribute__((ext_vector_type(8)));

#define WMMA_F32(a, b, c) \
  __builtin_amdgcn_wmma_f32_16x16x4_f32(false, (a), false, (b), (short)0, (c), false, false)

// Shapes
#define BB 2
#define NN 256
#define EE 768
#define HH 12
#define DD 64
#define LOG_2PI 1.8378770664093453f

// ---------------------------------------------------------------------------
// 1) log_softmax(mix_coef) -> logmix[768]
// ---------------------------------------------------------------------------
__global__ void logmix_kernel(const float* __restrict__ mix, float* __restrict__ logmix) {
  __shared__ float red[256];
  const int t = threadIdx.x;
  float m = -3.0e38f;
  for (int j = t; j < EE; j += 256) m = fmaxf(m, mix[j]);
  red[t] = m; __syncthreads();
  for (int s = 128; s > 0; s >>= 1) { if (t < s) red[t] = fmaxf(red[t], red[t + s]); __syncthreads(); }
  const float mx = red[0];
  __syncthreads();
  float sum = 0.f;
  for (int j = t; j < EE; j += 256) sum += __expf(mix[j] - mx);
  red[t] = sum; __syncthreads();
  for (int s = 128; s > 0; s >>= 1) { if (t < s) red[t] += red[t + s]; __syncthreads(); }
  const float lse = mx + __logf(red[0]);
  for (int j = t; j < EE; j += 256) logmix[j] = mix[j] - lse;
}

// ---------------------------------------------------------------------------
// 2) Mixture constants packed as float4 {A, B, C, 0} per (i,j):
//    t(k) = A + B*k + C*k^2  ==  log_mix[j] + logN(k; mu, sigma)
// ---------------------------------------------------------------------------
__global__ void mixconst_kernel(const float* __restrict__ mean, const float* __restrict__ logvar,
                                const float* __restrict__ logmix, v4f* __restrict__ ABC) {
  const int idx = blockIdx.x * 256 + threadIdx.x;   // < 768*768
  const float lv  = logvar[idx];
  const float mu  = mean[idx];
  const float inv = __expf(-lv);                    // 1/sigma^2
  const int j = idx % EE;
  v4f o;
  o.x = logmix[j] - 0.5f * LOG_2PI - 0.5f * lv - 0.5f * mu * mu * inv;
  o.y = mu * inv;
  o.z = -0.5f * inv;
  o.w = 0.f;
  ABC[idx] = o;
}

// ---------------------------------------------------------------------------
// 3) Pack weight K-row pairs: Wp[p][n] = {W[2p][n], W[2p+1][n]}
//    so a WMMA B fragment is one b64 load.
// ---------------------------------------------------------------------------
__global__ void pack_pairs_kernel(const float* __restrict__ W, v2f* __restrict__ Wp, int ldn) {
  const int n = blockIdx.x * 256 + threadIdx.x;
  const int p = blockIdx.y;
  v2f v = { W[(size_t)(2 * p) * ldn + n], W[(size_t)(2 * p + 1) * ldn + n] };
  Wp[(size_t)p * ldn + n] = v;
}

// ---------------------------------------------------------------------------
// 4) QKV GEMM: X(512x768) @ Wqkv(768x2304,pair-packed) + bqkv, de-interleave.
//    Block = 256 thr = 8 waves; wave tile 16Mx64N; block tile 32Mx256N.
//    V is stored pair-packed (rows n paired) for the att@V WMMA B side.
// ---------------------------------------------------------------------------
__device__ __forceinline__ void qkv_store(v8f acc, int n, int m0, int g,
                                          const float* __restrict__ bias,
                                          float* __restrict__ Q, float* __restrict__ Kk,
                                          float* __restrict__ Vp) {
  const int h = n / 192, rem = n % 192;
  const int d = rem / 3, s = rem - 3 * d;
  const float bv = bias[n];
#pragma unroll
  for (int r = 0; r < 8; r++) {
    const int m = m0 + 8 * g + r;
    const int b = m >> 8, nn = m & 255;
    const float val = acc[r] + bv;
    if (s == 0)      Q[((size_t)((b * HH + h) * NN + nn)) * DD + d] = val;
    else if (s == 1) Kk[((size_t)(b * NN + nn)) * EE + h * DD + d] = val;
    else             Vp[((size_t)(b * HH + h) * 128 + (nn >> 1)) * 128 + 2 * d + (nn & 1)] = val;
  }
}

__global__ void qkv_gemm_kernel(const float* __restrict__ X, const v2f* __restrict__ Wp,
                                const float* __restrict__ bias,
                                float* __restrict__ Q, float* __restrict__ Kk,
                                float* __restrict__ Vp) {
  const int lane = threadIdx.x & 31;
  const int w    = threadIdx.x >> 5;
  const int lo   = lane & 15;
  const int g    = lane >> 4;
  const int m0   = blockIdx.x * 32 + (w & 1) * 16;
  const int n0   = blockIdx.y * 256 + (w >> 1) * 64;
  v8f a0 = {0}, a1 = {0}, a2 = {0}, a3 = {0};
  const float* arow = X + (size_t)(m0 + lo) * EE;
#pragma unroll 2
  for (int kk = 0; kk < EE; kk += 4) {
    const int p = (kk >> 1) + g;                       // pair row
    const v2f a = *(const v2f*)(arow + kk + 2 * g);
    const v2f* bp = Wp + (size_t)p * 2304;
    const int pnext = (p + 8 < 384) ? (p + 8) : p;     // prefetch 4 K-steps ahead
    __builtin_prefetch((const void*)(Wp + (size_t)pnext * 2304 + n0 + lo), 0, 1);
    v2f b0 = bp[n0 + lo];
    v2f b1 = bp[n0 + 16 + lo];
    v2f b2 = bp[n0 + 32 + lo];
    v2f b3 = bp[n0 + 48 + lo];
    a0 = WMMA_F32(a, b0, a0);
    a1 = WMMA_F32(a, b1, a1);
    a2 = WMMA_F32(a, b2, a2);
    a3 = WMMA_F32(a, b3, a3);
  }
  qkv_store(a0, n0 + lo,      m0, g, bias, Q, Kk, Vp);
  qkv_store(a1, n0 + 16 + lo, m0, g, bias, Q, Kk, Vp);
  qkv_store(a2, n0 + 32 + lo, m0, g, bias, Q, Kk, Vp);
  qkv_store(a3, n0 + 48 + lo, m0, g, bias, Q, Kk, Vp);
}

// ---------------------------------------------------------------------------
// 5) Mixture logsumexp. Block owns one i (768 float4 coeffs in LDS, b128 ops),
//    256 bn values. KM output in (b,h,n,d) layout for attention.
// ---------------------------------------------------------------------------
__global__ void mixture_kernel(const float* __restrict__ keys, const v4f* __restrict__ ABC,
                               float* __restrict__ KM) {
  __shared__ v4f sABC[EE];    // 12 KB
  const int i  = blockIdx.x >> 1;
  const int bn = (blockIdx.x & 1) * 256 + threadIdx.x;
  const size_t rb = (size_t)i * EE;
  for (int j = threadIdx.x; j < EE; j += 256) sABC[j] = ABC[rb + j];
  __syncthreads();
  const float k = keys[(size_t)bn * EE + i];
  float mx = -3.0e38f;
#pragma unroll 4
  for (int j = 0; j < EE; j++) {
    const v4f c = sABC[j];
    const float t = fmaf(k, fmaf(k, c.z, c.y), c.x);
    mx = fmaxf(mx, t);
  }
  float s = 0.f;
#pragma unroll 4
  for (int j = 0; j < EE; j++) {
    const v4f c = sABC[j];
    const float t = fmaf(k, fmaf(k, c.z, c.y), c.x);
    s += __expf(t - mx);
  }
  const float out = mx + __logf(s);
  const int b = bn >> 8, nn = bn & 255, h = i >> 6, d = i & 63;
  KM[((size_t)((b * HH + h) * NN + nn)) * DD + d] = out;
}

// ---------------------------------------------------------------------------
// 6) Attention: per (b,h,row-chunk of 64): E = Q@KM^T -> LDS, softmax/sqrt(E),
//    O = att@V -> attn buffer in (b,n,e) layout. 64KB LDS score strip.
// ---------------------------------------------------------------------------
__global__ void attention_kernel(const float* __restrict__ Q, const float* __restrict__ KM,
                                 const float* __restrict__ Vp, float* __restrict__ attn) {
  __shared__ float sm[64 * 256];
  const int lane = threadIdx.x & 31;
  const int w    = threadIdx.x >> 5;
  const int lo   = lane & 15, g = lane >> 4;
  const int chunk = blockIdx.x & 3;
  const int bh    = blockIdx.x >> 2;
  const int b = bh / HH, h = bh % HH;
  const float* Qb  = Q  + (size_t)bh * NN * DD;
  const float* Kb  = KM + (size_t)bh * NN * DD;
  const float* Vbp = Vp + (size_t)bh * 128 * 128;   // pair-packed V

  // Phase 1: energy strip (64 x 256) into LDS via WMMA
  for (int tile = w; tile < 64; tile += 8) {
    const int mt = tile >> 4, nt = tile & 15;
    const float* qr = Qb + (size_t)(chunk * 64 + mt * 16 + lo) * DD;
    const float* kr = Kb + (size_t)(nt * 16 + lo) * DD;
    v8f acc = {0};
#pragma unroll
    for (int kk = 0; kk < DD; kk += 4) {
      const v2f a  = *(const v2f*)(qr + kk + 2 * g);
      const v2f bf = *(const v2f*)(kr + kk + 2 * g);
      acc = WMMA_F32(a, bf, acc);
    }
#pragma unroll
    for (int r = 0; r < 8; r++) sm[(mt * 16 + r + 8 * g) * 256 + nt * 16 + lo] = acc[r];
  }
  __syncthreads();

  // Phase 2: row softmax, then scale by 1/sqrt(768) (reference scales AFTER softmax)
  for (int row = w * 8; row < w * 8 + 8; row++) {
    float* rp = sm + row * 256;
    float vv[8];
    float mx = -3.0e38f;
#pragma unroll
    for (int j = 0; j < 8; j++) { vv[j] = rp[lane + 32 * j]; mx = fmaxf(mx, vv[j]); }
#pragma unroll
    for (int off = 16; off > 0; off >>= 1) mx = fmaxf(mx, __shfl_xor(mx, off, 32));
    float s = 0.f;
#pragma unroll
    for (int j = 0; j < 8; j++) { vv[j] = __expf(vv[j] - mx); s += vv[j]; }
#pragma unroll
    for (int off = 16; off > 0; off >>= 1) s += __shfl_xor(s, off, 32);
    const float scale = 1.f / (s * 27.712812921102035f);  // sqrt(768)
#pragma unroll
    for (int j = 0; j < 8; j++) rp[lane + 32 * j] = vv[j] * scale;
  }
  __syncthreads();

  // Phase 3: O(64x64) = att(64x256, LDS) @ V(256x64, pair-packed); store (b,n,e)
  for (int tile = w * 2; tile < w * 2 + 2; tile++) {
    const int mt = tile >> 2, nt = tile & 3;
    v8f acc = {0};
    const float* arow = sm + (mt * 16 + lo) * 256;
#pragma unroll 4
    for (int kk = 0; kk < 256; kk += 4) {
      const int p = (kk >> 1) + g;                  // V row pair (kk+2g, kk+2g+1)
      const v2f a  = *(const v2f*)(arow + kk + 2 * g);
      const v2f bf = *(const v2f*)(Vbp + (size_t)p * 128 + (nt * 16 + lo) * 2);
      acc = WMMA_F32(a, bf, acc);
    }
#pragma unroll
    for (int r = 0; r < 8; r++) {
      const int nrow = chunk * 64 + mt * 16 + r + 8 * g;
      attn[((size_t)(b * NN + nrow)) * EE + h * DD + nt * 16 + lo] = acc[r];
    }
  }
}

// ---------------------------------------------------------------------------
// 7) Projection GEMM: attn(512x768) @ Wproj(768x768,pair-packed) + bproj -> out
// ---------------------------------------------------------------------------
__global__ void proj_gemm_kernel(const float* __restrict__ X, const v2f* __restrict__ Wp,
                                 const float* __restrict__ bias, float* __restrict__ out) {
  const int lane = threadIdx.x & 31;
  const int w    = threadIdx.x >> 5;
  const int lo   = lane & 15, g = lane >> 4;
  const int m0   = blockIdx.x * 32 + (w & 1) * 16;
  const int n0   = blockIdx.y * 256 + (w >> 1) * 64;
  v8f accs[4] = {{0}, {0}, {0}, {0}};
  const float* arow = X + (size_t)(m0 + lo) * EE;
#pragma unroll 2
  for (int kk = 0; kk < EE; kk += 4) {
    const int p = (kk >> 1) + g;
    const v2f a = *(const v2f*)(arow + kk + 2 * g);
    const v2f* bp = Wp + (size_t)p * EE;
    const int pnext = (p + 8 < 384) ? (p + 8) : p;
    __builtin_prefetch((const void*)(Wp + (size_t)pnext * EE + n0 + lo), 0, 1);
    v2f b0 = bp[n0 + lo];
    v2f b1 = bp[n0 + 16 + lo];
    v2f b2 = bp[n0 + 32 + lo];
    v2f b3 = bp[n0 + 48 + lo];
    accs[0] = WMMA_F32(a, b0, accs[0]);
    accs[1] = WMMA_F32(a, b1, accs[1]);
    accs[2] = WMMA_F32(a, b2, accs[2]);
    accs[3] = WMMA_F32(a, b3, accs[3]);
  }
#pragma unroll
  for (int t = 0; t < 4; t++) {
    const int n = n0 + t * 16 + lo;
    const float bv = bias[n];
#pragma unroll
    for (int r = 0; r < 8; r++) {
      const int m = m0 + 8 * g + r;
      out[(size_t)m * EE + n] = accs[t][r] + bv;
    }
  }
}

// ---------------------------------------------------------------------------
extern "C" void kernel_launch(void* const* d_in, const int* in_sizes, int n_in,
                              void* d_out, int out_size, void* d_ws, size_t ws_size,
                              hipStream_t stream) {
  (void)in_sizes; (void)n_in; (void)out_size; (void)ws_size;
  const float* x      = (const float*)d_in[0];
  const float* Wqkv   = (const float*)d_in[1];
  const float* bqkv   = (const float*)d_in[2];
  const float* Wproj  = (const float*)d_in[3];
  const float* bproj  = (const float*)d_in[4];
  const float* mean   = (const float*)d_in[5];
  const float* logvar = (const float*)d_in[6];
  const float* mix    = (const float*)d_in[7];

  float* ws     = (float*)d_ws;
  float* logmix = ws;                         // 768
  v4f*   ABC    = (v4f*)(ws + 768);           // 768*768 float4 (16B aligned)
  float* base   = ws + 768 + 768 * 768 * 4;
  float* Qb     = base;                       // B*H*N*D = 393216 each below
  float* Kb     = Qb + BB * EE * NN;
  float* Vpb    = Kb + BB * EE * NN;          // pair-packed V
  float* KMb    = Vpb + BB * EE * NN;
  float* attnb  = KMb + BB * EE * NN;
  v2f*   WqkvP  = (v2f*)(attnb + BB * EE * NN);      // 384*2304 float2
  v2f*   WprojP = (v2f*)((float*)WqkvP + 768 * 2304); // 384*768 float2
  float* out    = (float*)d_out;

  logmix_kernel<<<1, 256, 0, stream>>>(mix, logmix);
  mixconst_kernel<<<(768 * 768) / 256, 256, 0, stream>>>(mean, logvar, logmix, ABC);
  pack_pairs_kernel<<<dim3(2304 / 256, 384), 256, 0, stream>>>(Wqkv, WqkvP, 2304);
  pack_pairs_kernel<<<dim3(768 / 256, 384), 256, 0, stream>>>(Wproj, WprojP, 768);
  qkv_gemm_kernel<<<dim3(512 / 32, 2304 / 256), 256, 0, stream>>>(x, WqkvP, bqkv, Qb, Kb, Vpb);
  mixture_kernel<<<768 * 2, 256, 0, stream>>>(Kb, ABC, KMb);
  attention_kernel<<<BB * HH * 4, 256, 0, stream>>>(Qb, KMb, Vpb, attnb);
  proj_gemm_kernel<<<dim3(512 / 32, 768 / 256), 256, 0, stream>>>(attnb, WprojP, bproj, out);
}